// LOGO_55095840473292
// MI455X (gfx1250) — compile-verified
//
#include <hip/hip_runtime.h>
#include <math.h>

typedef __attribute__((ext_vector_type(2))) float v2f;
typedef __attribute__((ext_vector_type(8))) float v8f;

// ---------------------------------------------------------------------------
// fp32 WMMA: D(16x16) = A(16x4) * B(4x16) + C   (wave32, gfx1250)
// A frag : lane m = lane&15, holds K = (lane>>4)*2 + {0,1}
// B frag : lane n = lane&15, holds K = (lane>>4)*2 + {0,1}
// C/D    : lane n = lane&15, rows m = (lane>>4)*8 + i  (i = vgpr 0..7)
// ---------------------------------------------------------------------------
__device__ __forceinline__ v8f wmma4(v2f a, v2f b, v8f c) {
  return __builtin_amdgcn_wmma_f32_16x16x4_f32(false, a, false, b, (short)0, c,
                                               false, false);
}

// Async global->LDS copy (gfx1250 GLOBAL_LOAD_ASYNC_TO_LDS_B128, ASYNCcnt).
// ldsoff = LDS byte offset (low 32 bits of a flat shared pointer).
__device__ __forceinline__ void async_g2l_b128(unsigned ldsoff,
                                               const float* gaddr) {
  asm volatile("global_load_async_to_lds_b128 %0, %1, off"
               :
               : "v"(ldsoff), "v"(gaddr)
               : "memory");
}
__device__ __forceinline__ void wait_async0() {
  asm volatile("s_wait_asynccnt 0x0" ::: "memory");
}

// ---------------------------------------------------------------------------
// Depthwise cross-correlation: x (B*C,31,31) (*) z (B*C,7,7) -> (B*C,25,25)
// ---------------------------------------------------------------------------
__global__ void k_xcorr(const float* __restrict__ x, const float* __restrict__ z,
                        float* __restrict__ out, int total) {
  int i = blockIdx.x * blockDim.x + threadIdx.x;
  if (i >= total) return;
  int p = i % 625;
  int bc = i / 625;
  int oy = p / 25, ox = p - oy * 25;
  const float* xp = x + (size_t)bc * 961 + oy * 31 + ox;
  const float* zp = z + (size_t)bc * 49;
  float s = 0.f;
#pragma unroll
  for (int r = 0; r < 7; ++r)
#pragma unroll
    for (int c = 0; c < 7; ++c) s += xp[r * 31 + c] * zp[r * 7 + c];
  out[i] = s;
}

// ---------------------------------------------------------------------------
// Implicit-im2col gather for 3x3 conv, pad 1.
// Unconditional load from a clamped (always-legal) address, then value-select.
// ---------------------------------------------------------------------------
__device__ __forceinline__ float conv_fetch(const float* __restrict__ Xb,
                                            int kk, int iy0, int ix0, int IW) {
  int c = kk / 9;
  int rs = kk - c * 9;
  int r = rs / 3;
  int s = rs - r * 3;
  int iy = iy0 + r, ix = ix0 + s;
  bool ok = ((unsigned)iy < (unsigned)IW) && ((unsigned)ix < (unsigned)IW);
  int iyc = min(max(iy, 0), IW - 1);
  int ixc = min(max(ix, 0), IW - 1);
  float v = Xb[((size_t)c * IW + iyc) * IW + ixc];
  return ok ? v : 0.f;
}

// ---------------------------------------------------------------------------
// Pipelined WMMA GEMM with async global->LDS weight staging.
//   C[M,N] = A[M,K] @ W[N,K]^T (+ epilogue)
// Block = 4 waves sharing one 64-wide N strip over a 64-row M block.
// Weight slab (64n x 32k fp32, 8KB) double-buffered in LDS, filled by
// GLOBAL_LOAD_ASYNC_TO_LDS_B128 one slab ahead (ASYNCcnt + barrier per slab).
// A fragments prefetched into registers one slab ahead (LOADcnt).
// AMODE 0: dense A (row-major, lda=K); epilogue bias(+ReLU); out row-major.
// AMODE 1: implicit im2col (3x3 pad1 stride 1|2, square IWxIW in, 13x13 out,
//          m = b*169+p, K = Cin*9); epilogue BN+ReLU (bn_g!=0) or +bias;
//          out NCHW (B,N,13,13).
// Requires: M%64==0, N%64==0, K%32==0.  Grid = (M/64)*(N/64), block = 128.
// ---------------------------------------------------------------------------
template <int AMODE>
__global__ __launch_bounds__(128) void k_gemm(
    const float* __restrict__ A, const float* __restrict__ W,
    const float* __restrict__ bias, const float* __restrict__ bn_g,
    const float* __restrict__ bn_b, const float* __restrict__ bn_m,
    const float* __restrict__ bn_v, float* __restrict__ O, int M, int N, int K,
    int relu, int Cin, int IW, int stride) {
  const int KS = 32;  // k-slab
  const int LW = 36;  // padded LDS row stride (words)
  __shared__ float lw[2][64 * LW];

  int tid = threadIdx.x;
  int lane = tid & 31, wave = tid >> 5;
  int mblocks = M >> 6;
  int mblk = blockIdx.x % mblocks;
  int nstrip = blockIdx.x / mblocks;
  int n0 = nstrip << 6;
  int m0 = ((mblk << 2) + wave) << 4;
  int mi = lane & 15, koff = (lane >> 4) << 1;

  // ---- A-side addressing ----
  int m = m0 + mi;
  const float* Ar = nullptr;
  const float* Xb = nullptr;
  int iy0 = 0, ix0 = 0;
  if (AMODE == 0) {
    Ar = A + (size_t)m * K + koff;
  } else {
    int b = m / 169;
    int p = m - b * 169;
    int oy = p / 13, ox = p - oy * 13;
    iy0 = oy * stride - 1;
    ix0 = ox * stride - 1;
    Xb = A + (size_t)b * Cin * IW * IW;
  }

  // ---- W staging map: 128 threads x 4 x (b128) = 64n x 32k slab ----
  int wn[4], wkc[4];
  unsigned lbase[2][4];
#pragma unroll
  for (int i = 0; i < 4; ++i) {
    int e = (i * 128 + tid) << 2;
    wn[i] = e >> 5;
    wkc[i] = e & 31;
    lbase[0][i] = (unsigned)(size_t)&lw[0][wn[i] * LW + wkc[i]];
    lbase[1][i] = (unsigned)(size_t)&lw[1][wn[i] * LW + wkc[i]];
  }

  int nslab = K / KS;

  auto asyncW = [&](int buf, int k0) {
#pragma unroll
    for (int i = 0; i < 4; ++i)
      async_g2l_b128(lbase[buf][i],
                     W + (size_t)(n0 + wn[i]) * K + k0 + wkc[i]);
  };
  auto loadA = [&](int k0, v2f* ar) {
#pragma unroll
    for (int s = 0; s < 8; ++s) {
      if (AMODE == 0) {
        ar[s] = *(const v2f*)(Ar + k0 + s * 4);
      } else {
        int kk = k0 + s * 4 + koff;
        v2f a;
        a.x = conv_fetch(Xb, kk, iy0, ix0, IW);
        a.y = conv_fetch(Xb, kk + 1, iy0, ix0, IW);
        ar[s] = a;
      }
    }
  };

  v8f acc0 = {}, acc1 = {}, acc2 = {}, acc3 = {};
  v2f aCur[8], aNext[8];

  asyncW(0, 0);
  loadA(0, aCur);
  wait_async0();
  __syncthreads();

  for (int sl = 0; sl < nslab; ++sl) {
    int buf = sl & 1;
    if (sl + 1 < nslab) {
      asyncW(buf ^ 1, (sl + 1) * KS);  // DMA in flight during compute
      loadA((sl + 1) * KS, aNext);     // A prefetch in flight during compute
    }
    const float* lb = &lw[buf][0];
#pragma unroll
    for (int s = 0; s < 8; ++s) {
      int kw = s * 4 + koff;
      v2f b0 = *(const v2f*)&lb[(mi) * LW + kw];
      v2f b1 = *(const v2f*)&lb[(16 + mi) * LW + kw];
      v2f b2 = *(const v2f*)&lb[(32 + mi) * LW + kw];
      v2f b3 = *(const v2f*)&lb[(48 + mi) * LW + kw];
      acc0 = wmma4(aCur[s], b0, acc0);
      acc1 = wmma4(aCur[s], b1, acc1);
      acc2 = wmma4(aCur[s], b2, acc2);
      acc3 = wmma4(aCur[s], b3, acc3);
    }
    if (sl + 1 < nslab) {
      wait_async0();
      __syncthreads();
#pragma unroll
      for (int s = 0; s < 8; ++s) aCur[s] = aNext[s];
    }
  }

  // ---- epilogue ----
  int mbase = m0 + ((lane >> 4) << 3);
#pragma unroll
  for (int j = 0; j < 4; ++j) {
    v8f acc = (j == 0) ? acc0 : (j == 1) ? acc1 : (j == 2) ? acc2 : acc3;
    int n = n0 + (j << 4) + mi;
    if (AMODE == 0) {
      float bb = bias ? bias[n] : 0.f;
#pragma unroll
      for (int i = 0; i < 8; ++i) {
        float v = acc[i] + bb;
        if (relu) v = fmaxf(v, 0.f);
        O[(size_t)(mbase + i) * N + n] = v;
      }
    } else {
      float scale, shift;
      int dorelu;
      if (bn_g) {
        float sc = bn_g[n] * rsqrtf(bn_v[n] + 1e-5f);
        scale = sc;
        shift = bn_b[n] - bn_m[n] * sc;
        dorelu = 1;
      } else {
        scale = 1.f;
        shift = bias[n];
        dorelu = 0;
      }
#pragma unroll
      for (int i = 0; i < 8; ++i) {
        int mm = mbase + i;
        int bb2 = mm / 169;
        int pp = mm - bb2 * 169;
        float v = acc[i] * scale + shift;
        if (dorelu) v = fmaxf(v, 0.f);
        O[((size_t)bb2 * N + n) * 169 + pp] = v;
      }
    }
  }
}

// ---------------------------------------------------------------------------
// Attention scores: S[b,h,nq,nk] = scale * sum_d Q[nq,b,h,d]*K[nk,b,h,d] + mask
// Q/K layout: [(n*64+b)*192 + h*24 + d].  One wave = one 16x16 tile.
// ---------------------------------------------------------------------------
__global__ __launch_bounds__(128) void k_scores(const float* __restrict__ Q,
                                                const float* __restrict__ Kb,
                                                float* __restrict__ S,
                                                const int* __restrict__ msz) {
  int wid = blockIdx.x * (blockDim.x >> 5) + (threadIdx.x >> 5);
  int lane = threadIdx.x & 31;
  int tn = wid % 11;
  int t2 = wid / 11;
  int tm = t2 % 11;
  int bh = t2 / 11;
  if (bh >= 512) return;
  int b = bh >> 3, h = bh & 7;
  int m0 = tm << 4, n0 = tn << 4;
  int mi = lane & 15, koff = (lane >> 4) << 1;

  int nq = m0 + mi;
  int nk = n0 + mi;
  int nqc = nq < 169 ? nq : 168;
  int nkc = nk < 169 ? nk : 168;
  const float* Qp = Q + ((size_t)nqc * 64 + b) * 192 + h * 24 + koff;
  const float* Kp = Kb + ((size_t)nkc * 64 + b) * 192 + h * 24 + koff;

  v8f acc = {};
#pragma unroll
  for (int k = 0; k < 24; k += 4) {
    v2f a = *(const v2f*)(Qp + k);
    v2f bb = *(const v2f*)(Kp + k);
    acc = wmma4(a, bb, acc);
  }

  if (nk >= 169) return;
  int ms2 = msz[0] >> 1;
  int rk = nk / 13, ck = nk - rk * 13;
  float* Sp = S + (size_t)bh * 28561;
  int mbase = m0 + ((lane >> 4) << 3);
  const float scale = 0.20412414523193154f;  // 24^-0.5
#pragma unroll
  for (int i = 0; i < 8; ++i) {
    int q2 = mbase + i;
    if (q2 < 169) {
      int rq = q2 / 13, cq = q2 - rq * 13;
      int allowed = (abs(rq - rk) <= ms2) && (abs(cq - ck) <= ms2);
      Sp[(size_t)q2 * 169 + nk] = acc[i] * scale + (allowed ? 0.f : -1e9f);
    }
  }
}

// ---------------------------------------------------------------------------
// Row softmax over 169 columns (mask bias already included).
// ---------------------------------------------------------------------------
__global__ void k_softmax(float* __restrict__ S, int rows) {
  int row = blockIdx.x * (blockDim.x >> 5) + (threadIdx.x >> 5);
  if (row >= rows) return;
  int lane = threadIdx.x & 31;
  float* r = S + (size_t)row * 169;
  float mx = -3.4e38f;
  for (int j = lane; j < 169; j += 32) mx = fmaxf(mx, r[j]);
#pragma unroll
  for (int o = 16; o > 0; o >>= 1) mx = fmaxf(mx, __shfl_xor(mx, o, 32));
  float sum = 0.f;
  for (int j = lane; j < 169; j += 32) {
    float e = __expf(r[j] - mx);
    r[j] = e;
    sum += e;
  }
#pragma unroll
  for (int o = 16; o > 0; o >>= 1) sum += __shfl_xor(sum, o, 32);
  float inv = 1.f / sum;
  for (int j = lane; j < 169; j += 32) r[j] *= inv;
}

// ---------------------------------------------------------------------------
// O[nq,b,h,d] = sum_k S[b,h,nq,k] * V[k,b,h,d].  Wave tile: 16(M) x 32(N=d).
// K-edge handled by value-selects after unconditional clamped-address loads.
// ---------------------------------------------------------------------------
__global__ __launch_bounds__(128) void k_av(const float* __restrict__ S,
                                            const float* __restrict__ V,
                                            float* __restrict__ O) {
  int wid = blockIdx.x * (blockDim.x >> 5) + (threadIdx.x >> 5);
  int lane = threadIdx.x & 31;
  int tm = wid % 11;
  int bh = wid / 11;
  if (bh >= 512) return;
  int b = bh >> 3, h = bh & 7;
  int m0 = tm << 4;
  int mi = lane & 15, koff = (lane >> 4) << 1;

  int nq = m0 + mi;
  int nqc = nq < 169 ? nq : 168;
  const float* Sp = S + (size_t)bh * 28561 + (size_t)nqc * 169;
  int d1 = 16 + mi;
  int d1c = d1 < 24 ? d1 : 23;
  int d1ok = d1 < 24;

  v8f a0 = {}, a1 = {};
  for (int k = 0; k < 169; k += 4) {
    int kk = k + koff;
    int kv0 = kk < 169, kv1 = (kk + 1) < 169;
    int kc0 = kv0 ? kk : 168;
    int kc1 = kv1 ? kk + 1 : 168;
    // unconditional loads (addresses legal), masked by value selects
    float ax = Sp[kk];  // may overrun row by <=3 floats, still in buffer
    float ay = Sp[kk + 1];
    v2f a;
    a.x = kv0 ? ax : 0.f;
    a.y = kv1 ? ay : 0.f;
    const float* V0 = V + ((size_t)kc0 * 64 + b) * 192 + h * 24;
    const float* V1 = V + ((size_t)kc1 * 64 + b) * 192 + h * 24;
    float v00 = V0[mi], v10 = V1[mi];
    float v01 = V0[d1c], v11 = V1[d1c];
    v2f b0, b1;
    b0.x = kv0 ? v00 : 0.f;
    b0.y = kv1 ? v10 : 0.f;
    b1.x = (kv0 && d1ok) ? v01 : 0.f;
    b1.y = (kv1 && d1ok) ? v11 : 0.f;
    a0 = wmma4(a, b0, a0);
    a1 = wmma4(a, b1, a1);
  }

  int mbase = m0 + ((lane >> 4) << 3);
#pragma unroll
  for (int i = 0; i < 8; ++i) {
    int n = mbase + i;
    if (n >= 169) continue;
    float* Op = O + ((size_t)n * 64 + b) * 192 + h * 24;
    Op[mi] = a0[i];
    if (d1ok) Op[d1] = a1[i];
  }
}

// ---------------------------------------------------------------------------
// out = LayerNorm(X + Y) * g + b over last dim (192).  One wave per row.
// ---------------------------------------------------------------------------
__global__ void k_add_ln(const float* __restrict__ X, const float* __restrict__ Y,
                         const float* __restrict__ g, const float* __restrict__ bb,
                         float* __restrict__ O, int M) {
  int row = blockIdx.x * (blockDim.x >> 5) + (threadIdx.x >> 5);
  if (row >= M) return;
  int lane = threadIdx.x & 31;
  const float* xr = X + (size_t)row * 192;
  const float* yr = Y + (size_t)row * 192;
  float vals[6];
  float s = 0.f, s2 = 0.f;
#pragma unroll
  for (int t = 0; t < 6; ++t) {
    int c = lane + t * 32;
    float v = xr[c] + yr[c];
    vals[t] = v;
    s += v;
    s2 += v * v;
  }
#pragma unroll
  for (int o = 16; o > 0; o >>= 1) {
    s += __shfl_xor(s, o, 32);
    s2 += __shfl_xor(s2, o, 32);
  }
  float mu = s * (1.f / 192.f);
  float var = s2 * (1.f / 192.f) - mu * mu;
  float inv = rsqrtf(var + 1e-5f);
  float* orow = O + (size_t)row * 192;
#pragma unroll
  for (int t = 0; t < 6; ++t) {
    int c = lane + t * 32;
    orow[c] = (vals[t] - mu) * inv * g[c] + bb[c];
  }
}

// ---------------------------------------------------------------------------
// (B,192,169) NCHW  <->  seq [(p*64+b)*192 + c]
// ---------------------------------------------------------------------------
__global__ void k_pack(const float* __restrict__ R, float* __restrict__ Sq,
                       int total) {
  int i = blockIdx.x * blockDim.x + threadIdx.x;
  if (i >= total) return;
  int p = i % 169;
  int t = i / 169;
  int c = t % 192;
  int b = t / 192;
  Sq[((size_t)p * 64 + b) * 192 + c] = R[i];
}

__global__ void k_unpack(const float* __restrict__ Sq, float* __restrict__ R,
                         int total) {
  int i = blockIdx.x * blockDim.x + threadIdx.x;
  if (i >= total) return;
  int p = i % 169;
  int t = i / 169;
  int c = t % 192;
  int b = t / 192;
  R[i] = Sq[((size_t)p * 64 + b) * 192 + c];
}

// ---------------------------------------------------------------------------
// O = X + pos,  pos[p][c] = c<96 ? col_embed[p%13][c] : row_embed[p/13][c-96]
// ---------------------------------------------------------------------------
__global__ void k_add_pos(const float* __restrict__ X, const float* __restrict__ ce,
                          const float* __restrict__ re, float* __restrict__ O,
                          int total) {
  int i = blockIdx.x * blockDim.x + threadIdx.x;
  if (i >= total) return;
  int c = i % 192;
  int m = i / 192;
  int p = m >> 6;  // m = p*64 + b
  float pos = (c < 96) ? ce[(p % 13) * 96 + c] : re[(p / 13) * 96 + (c - 96)];
  O[i] = X[i] + pos;
}

// ---------------------------------------------------------------------------
// GroupNorm(32 groups, C=192 -> 6 ch/group over 169 px) + ReLU.
// One wave per (b, group).  Layout (B,192,169).
// ---------------------------------------------------------------------------
__global__ void k_gn_relu(const float* __restrict__ X, const float* __restrict__ G,
                          const float* __restrict__ Bt, float* __restrict__ O) {
  int grp = blockIdx.x;  // 64*32
  int b = grp >> 5, g = grp & 31;
  int lane = threadIdx.x;
  const float* base = X + ((size_t)b * 192 + g * 6) * 169;
  float s = 0.f, s2 = 0.f;
  for (int i = lane; i < 1014; i += 32) {
    float v = base[i];
    s += v;
    s2 += v * v;
  }
#pragma unroll
  for (int o = 16; o > 0; o >>= 1) {
    s += __shfl_xor(s, o, 32);
    s2 += __shfl_xor(s2, o, 32);
  }
  float mu = s * (1.f / 1014.f);
  float var = s2 * (1.f / 1014.f) - mu * mu;
  float inv = rsqrtf(var + 1e-5f);
  float* out = O + ((size_t)b * 192 + g * 6) * 169;
  for (int i = lane; i < 1014; i += 32) {
    int c = g * 6 + i / 169;
    float v = (base[i] - mu) * inv * G[c] + Bt[c];
    out[i] = fmaxf(v, 0.f);
  }
}

// ---------------------------------------------------------------------------
// Small output conv: 3x3 s1 pad1, Cin=192, Co in {4,2,1}.  One thread/output.
// ---------------------------------------------------------------------------
__global__ void k_conv_out(const float* __restrict__ X, const float* __restrict__ W,
                           const float* __restrict__ B, float* __restrict__ O,
                           int Co, int total) {
  int i = blockIdx.x * blockDim.x + threadIdx.x;
  if (i >= total) return;
  int p = i % 169;
  int t = i / 169;
  int co = t % Co;
  int b = t / Co;
  int oy = p / 13, ox = p - oy * 13;
  float acc = B[co];
  const float* Xb = X + (size_t)b * 192 * 169;
  const float* Wc = W + (size_t)co * 192 * 9;
  for (int c = 0; c < 192; ++c) {
    const float* xc = Xb + c * 169;
    const float* wc = Wc + c * 9;
#pragma unroll
    for (int r = 0; r < 3; ++r) {
      int iy = oy - 1 + r;
      bool rok = (unsigned)iy < 13u;
      int iyc = min(max(iy, 0), 12);
#pragma unroll
      for (int s = 0; s < 3; ++s) {
        int ix = ox - 1 + s;
        bool ok = rok && ((unsigned)ix < 13u);
        int ixc = min(max(ix, 0), 12);
        float v = xc[iyc * 13 + ixc];
        acc += (ok ? v : 0.f) * wc[r * 3 + s];
      }
    }
  }
  O[i] = acc;
}

// ---------------------------------------------------------------------------
static inline int cdiv(int a, int b) { return (a + b - 1) / b; }

extern "C" void kernel_launch(void* const* d_in, const int* in_sizes, int n_in,
                              void* d_out, int out_size, void* d_ws,
                              size_t ws_size, hipStream_t stream) {
  (void)in_sizes; (void)n_in; (void)out_size; (void)ws_size;
  const float* x0 = (const float*)d_in[0];
  const float* x1 = (const float*)d_in[1];
  const float* x2 = (const float*)d_in[2];
  const float* z0 = (const float*)d_in[3];
  const float* z1 = (const float*)d_in[4];
  const float* z2 = (const float*)d_in[5];
  const float* conv1_w = (const float*)d_in[6];
  const float* conv3_w = (const float*)d_in[7];
  const float* conv2_w = (const float*)d_in[8];
  const float* bn1_g = (const float*)d_in[9];
  const float* bn1_b = (const float*)d_in[10];
  const float* bn1_m = (const float*)d_in[11];
  const float* bn1_v = (const float*)d_in[12];
  const float* bn2_g = (const float*)d_in[13];
  const float* bn2_b = (const float*)d_in[14];
  const float* bn2_m = (const float*)d_in[15];
  const float* bn2_v = (const float*)d_in[16];
  const float* bn3_g = (const float*)d_in[17];
  const float* bn3_b = (const float*)d_in[18];
  const float* bn3_m = (const float*)d_in[19];
  const float* bn3_v = (const float*)d_in[20];
  const float* row_embed = (const float*)d_in[21];
  const float* col_embed = (const float*)d_in[22];
  const float* t_wq = (const float*)d_in[23];
  const float* t_bq = (const float*)d_in[24];
  const float* t_wk = (const float*)d_in[25];
  const float* t_bk = (const float*)d_in[26];
  const float* t_wv = (const float*)d_in[27];
  const float* t_bv = (const float*)d_in[28];
  const float* t_wo = (const float*)d_in[29];
  const float* t_bo = (const float*)d_in[30];
  const float* t_ln1_g = (const float*)d_in[31];
  const float* t_ln1_b = (const float*)d_in[32];
  const float* t_ln2_g = (const float*)d_in[33];
  const float* t_ln2_b = (const float*)d_in[34];
  const float* t_ff1_w = (const float*)d_in[35];
  const float* t_ff1_b = (const float*)d_in[36];
  const float* t_ff2_w = (const float*)d_in[37];
  const float* t_ff2_b = (const float*)d_in[38];
  const float* convloc_w = (const float*)d_in[39];
  const float* convloc_b = (const float*)d_in[40];
  const float* convloc_gn_g = (const float*)d_in[41];
  const float* convloc_gn_b = (const float*)d_in[42];
  const float* loc_out_w = (const float*)d_in[43];
  const float* loc_out_b = (const float*)d_in[44];
  const float* convcls_w = (const float*)d_in[45];
  const float* convcls_b = (const float*)d_in[46];
  const float* convcls_gn_g = (const float*)d_in[47];
  const float* convcls_gn_b = (const float*)d_in[48];
  const float* cls1_w = (const float*)d_in[49];
  const float* cls1_b = (const float*)d_in[50];
  const float* cls2_w = (const float*)d_in[51];
  const float* cls2_b = (const float*)d_in[52];
  const int* mask_size = (const int*)d_in[53];

  const int M = 10816;      // 169 * 64
  const int SEQ = 2076672;  // 10816 * 192
  float* ws = (float*)d_ws;
  float* corr = ws;  ws += 15360000;  // 64*384*625 (also reused as S)
  float* res1 = ws;  ws += SEQ;
  float* res2 = ws;  ws += SEQ;
  float* res3 = ws;  ws += SEQ;
  float* mem1 = ws;  ws += SEQ;
  float* mem2 = ws;  ws += SEQ;
  float* outA = ws;  ws += SEQ;
  float* outB = ws;  ws += SEQ;
  float* qin  = ws;  ws += SEQ;
  float* kin  = ws;  ws += SEQ;
  float* Qb   = ws;  ws += SEQ;
  float* Kb   = ws;  ws += SEQ;
  float* Vb   = ws;  ws += SEQ;
  float* Ob   = ws;  ws += SEQ;
  float* t1   = ws;  ws += SEQ;
  float* hid  = ws;  ws += 8306688;  // 10816 * 768
  float* Sbuf = corr;                // 64*8*169*169 (< corr size)
  float* hpA = res2;                 // head ping/pong reuse
  float* hpB = res3;

  // grid = (M/64) * (N/64): N=192 -> 507, N=768 -> 2028
  const int G192 = 169 * 3;
  const int G768 = 169 * 12;

  // ---- Stage A+B: xcorr -> conv(3x3,s2)+BN+ReLU -> (B,192,13,13) ----------
  {
    int tot = 64 * 384 * 625;
    k_xcorr<<<cdiv(tot, 256), 256, 0, stream>>>(x0, z0, corr, tot);
    k_gemm<1><<<G192, 128, 0, stream>>>(corr, conv1_w, nullptr, bn1_g, bn1_b,
                                        bn1_m, bn1_v, res1, M, 192, 3456, 0,
                                        384, 25, 2);
    k_xcorr<<<cdiv(tot, 256), 256, 0, stream>>>(x1, z1, corr, tot);
    k_gemm<1><<<G192, 128, 0, stream>>>(corr, conv3_w, nullptr, bn3_g, bn3_b,
                                        bn3_m, bn3_v, res2, M, 192, 3456, 0,
                                        384, 25, 2);
    int tot2 = 64 * 256 * 625;
    k_xcorr<<<cdiv(tot2, 256), 256, 0, stream>>>(x2, z2, corr, tot2);
    k_gemm<1><<<G192, 128, 0, stream>>>(corr, conv2_w, nullptr, bn2_g, bn2_b,
                                        bn2_m, bn2_v, res3, M, 192, 2304, 0,
                                        256, 25, 2);
  }

  // ---- Pack to sequence layout [(p*64+b), c] ------------------------------
  k_pack<<<cdiv(SEQ, 256), 256, 0, stream>>>(res1, mem1, SEQ);
  k_pack<<<cdiv(SEQ, 256), 256, 0, stream>>>(res2, mem2, SEQ);
  k_pack<<<cdiv(SEQ, 256), 256, 0, stream>>>(res3, outA, SEQ);

  // ---- Transformer: 2 layers ---------------------------------------------
  for (int li = 0; li < 2; ++li) {
    const float* mem = (li == 0) ? mem1 : mem2;
    k_add_pos<<<cdiv(SEQ, 256), 256, 0, stream>>>(outA, col_embed, row_embed,
                                                  qin, SEQ);
    k_add_pos<<<cdiv(SEQ, 256), 256, 0, stream>>>(mem, col_embed, row_embed,
                                                  kin, SEQ);
    k_gemm<0><<<G192, 128, 0, stream>>>(qin, t_wq + li * 36864, t_bq + li * 192,
                                        nullptr, nullptr, nullptr, nullptr, Qb,
                                        M, 192, 192, 0, 0, 0, 0);
    k_gemm<0><<<G192, 128, 0, stream>>>(kin, t_wk + li * 36864, t_bk + li * 192,
                                        nullptr, nullptr, nullptr, nullptr, Kb,
                                        M, 192, 192, 0, 0, 0, 0);
    k_gemm<0><<<G192, 128, 0, stream>>>(mem, t_wv + li * 36864, t_bv + li * 192,
                                        nullptr, nullptr, nullptr, nullptr, Vb,
                                        M, 192, 192, 0, 0, 0, 0);
    k_scores<<<15488, 128, 0, stream>>>(Qb, Kb, Sbuf, mask_size);
    k_softmax<<<21632, 128, 0, stream>>>(Sbuf, 512 * 169);
    k_av<<<1408, 128, 0, stream>>>(Sbuf, Vb, Ob);
    k_gemm<0><<<G192, 128, 0, stream>>>(Ob, t_wo + li * 36864, t_bo + li * 192,
                                        nullptr, nullptr, nullptr, nullptr, t1,
                                        M, 192, 192, 0, 0, 0, 0);
    k_add_ln<<<2704, 128, 0, stream>>>(outA, t1, t_ln1_g + li * 192,
                                       t_ln1_b + li * 192, outB, M);
    k_gemm<0><<<G768, 128, 0, stream>>>(outB, t_ff1_w + li * 147456,
                                        t_ff1_b + li * 768, nullptr, nullptr,
                                        nullptr, nullptr, hid, M, 768, 192, 1,
                                        0, 0, 0);
    k_gemm<0><<<G192, 128, 0, stream>>>(hid, t_ff2_w + li * 147456,
                                        t_ff2_b + li * 192, nullptr, nullptr,
                                        nullptr, nullptr, t1, M, 192, 768, 0,
                                        0, 0, 0);
    k_add_ln<<<2704, 128, 0, stream>>>(outB, t1, t_ln2_g + li * 192,
                                       t_ln2_b + li * 192, outA, M);
  }

  // ---- Back to (B,192,13,13) ----------------------------------------------
  float* resF = res1;
  k_unpack<<<cdiv(SEQ, 256), 256, 0, stream>>>(outA, resF, SEQ);

  float* outp = (float*)d_out;

  // ---- loc tower ----------------------------------------------------------
  {
    const float* y = resF;
    for (int i = 0; i < 3; ++i) {
      k_gemm<1><<<G192, 128, 0, stream>>>(y, convloc_w + (size_t)i * 331776,
                                          convloc_b + i * 192, nullptr, nullptr,
                                          nullptr, nullptr, hpA, M, 192, 1728,
                                          0, 192, 13, 1);
      k_gn_relu<<<2048, 32, 0, stream>>>(hpA, convloc_gn_g + i * 192,
                                         convloc_gn_b + i * 192, hpB);
      y = hpB;
    }
    int tot = 64 * 4 * 169;
    k_conv_out<<<cdiv(tot, 256), 256, 0, stream>>>(hpB, loc_out_w, loc_out_b,
                                                   outp, 4, tot);
  }

  // ---- cls tower ----------------------------------------------------------
  {
    const float* y = resF;
    for (int i = 0; i < 3; ++i) {
      k_gemm<1><<<G192, 128, 0, stream>>>(y, convcls_w + (size_t)i * 331776,
                                          convcls_b + i * 192, nullptr, nullptr,
                                          nullptr, nullptr, hpA, M, 192, 1728,
                                          0, 192, 13, 1);
      k_gn_relu<<<2048, 32, 0, stream>>>(hpA, convcls_gn_g + i * 192,
                                         convcls_gn_b + i * 192, hpB);
      y = hpB;
    }
    int tot1 = 64 * 2 * 169;
    k_conv_out<<<cdiv(tot1, 256), 256, 0, stream>>>(hpB, cls1_w, cls1_b,
                                                    outp + 43264, 2, tot1);
    int tot2 = 64 * 1 * 169;
    k_conv_out<<<cdiv(tot2, 256), 256, 0, stream>>>(hpB, cls2_w, cls2_b,
                                                    outp + 64896, 1, tot2);
  }
}